// RNN_73615739453777
// MI455X (gfx1250) — compile-verified
//
#include <hip/hip_runtime.h>

// CDNA5 / gfx1250, wave32. Fully fused 2-layer tanh-RNN using
// V_WMMA_F32_16X16X4_F32 for all matvecs (computed transposed: W @ H^T).
// Branchless tanh/sigmoid via v_exp_f32 + v_rcp_f32; D->B operand relayout
// via v_permlanex16 (pure VALU, no LDS round-trip); x streamed as float4
// (2 timesteps per load, depth-2 software pipeline).

typedef __attribute__((ext_vector_type(2))) float v2f;
typedef __attribute__((ext_vector_type(8))) float v8f;

static __device__ __forceinline__ v8f wmma4(v2f a, v2f b, v8f c) {
  // D(16x16) = A(16x4) * B(4x16) + C ; all f32
  return __builtin_amdgcn_wmma_f32_16x16x4_f32(false, a, false, b,
                                               (short)0, c, false, false);
}

// lane <-> lane^16 half-swap as a single VALU op (identity lane selects).
static __device__ __forceinline__ float xor16(float v) {
  int i = __builtin_bit_cast(int, v);
  int r = __builtin_amdgcn_permlanex16(i, i, 0x76543210, 0xfedcba98,
                                       false, false);
  return __builtin_bit_cast(float, r);
}

// tanh(x) = 1 - 2/(exp2(x*2log2e)+1): mul, v_exp_f32, add, v_rcp_f32, fma.
// Branchless; saturates to +/-1 correctly (exp2->inf => rcp->0; exp2->0 => 1-2).
static __device__ __forceinline__ float tanh_fast(float x) {
  constexpr float TWO_LOG2E = 2.885390081777926815f;  // 2*log2(e)
  float e = __builtin_amdgcn_exp2f(x * TWO_LOG2E);
  float r = __builtin_amdgcn_rcpf(e + 1.0f);
  return __builtin_fmaf(-2.0f, r, 1.0f);
}

// sigmoid(z) = 1/(1+exp2(-z*log2e))
static __device__ __forceinline__ float sigmoid_fast(float z) {
  constexpr float LOG2E = 1.442695040888963407f;
  float e = __builtin_amdgcn_exp2f(-z * LOG2E);
  return __builtin_amdgcn_rcpf(e + 1.0f);
}

constexpr int BB = 2048, TT = 1024, II = 2, HH = 8;

__global__ __launch_bounds__(128, 1) void rnn2_fused(
    const float* __restrict__ x,
    const float* __restrict__ w_ih0, const float* __restrict__ w_hh0,
    const float* __restrict__ b_ih0, const float* __restrict__ b_hh0,
    const float* __restrict__ w_ih1, const float* __restrict__ w_hh1,
    const float* __restrict__ b_ih1, const float* __restrict__ b_hh1,
    const float* __restrict__ w_out, const float* __restrict__ b_out,
    float* __restrict__ out)
{
  const int lane = threadIdx.x & 31;
  const int wid  = threadIdx.x >> 5;
  const int tile = blockIdx.x * (blockDim.x >> 5) + wid;  // 16-batch tile
  const int b0   = tile * 16;
  const int m    = lane & 15;       // A-matrix row (h_out) / batch lane
  const int hi   = lane >> 4;       // 0 -> K=0,1 ; 1 -> K=2,3 (A layout)
  const int kb   = 2 * hi;

  // ---------------- A operands: weights, constant over time -------------
  // A layout (16x4 f32): lane m holds (K=2*hi, K=2*hi+1) in its 2 VGPRs.
  v2f a_ih0  = {0.f, 0.f};                     // W_ih0 (8x2), K=2..3 zero
  v2f a_hh0a = {0.f, 0.f}, a_hh0b = {0.f, 0.f};
  v2f a_ih1a = {0.f, 0.f}, a_ih1b = {0.f, 0.f};
  v2f a_hh1a = {0.f, 0.f}, a_hh1b = {0.f, 0.f};
  if (m < HH) {
    if (hi == 0) { a_ih0.x = w_ih0[m*II+0]; a_ih0.y = w_ih0[m*II+1]; }
    a_hh0a.x = w_hh0[m*HH + kb + 0];     a_hh0a.y = w_hh0[m*HH + kb + 1];
    a_hh0b.x = w_hh0[m*HH + 4 + kb + 0]; a_hh0b.y = w_hh0[m*HH + 4 + kb + 1];
    a_ih1a.x = w_ih1[m*HH + kb + 0];     a_ih1a.y = w_ih1[m*HH + kb + 1];
    a_ih1b.x = w_ih1[m*HH + 4 + kb + 0]; a_ih1b.y = w_ih1[m*HH + 4 + kb + 1];
    a_hh1a.x = w_hh1[m*HH + kb + 0];     a_hh1a.y = w_hh1[m*HH + kb + 1];
    a_hh1b.x = w_hh1[m*HH + 4 + kb + 0]; a_hh1b.y = w_hh1[m*HH + 4 + kb + 1];
  }

  // ------------- C seeds: biases broadcast along the batch (N) dim ------
  // D layout: VGPR r, lanes 0-15 => row (h_out) r, N = lane (batch).
  v8f cb0 = {};
  v8f cb1 = {};
  if (hi == 0) {
#pragma unroll
    for (int r = 0; r < 8; ++r) {
      cb0[r] = b_ih0[r] + b_hh0[r];
      cb1[r] = b_ih1[r] + b_hh1[r];
    }
  }

  // ------------- hidden states, stored directly in B-operand layout -----
  // B (4x16 f32) chunk: VGPR0 = rows K,K+1 ; VGPR1 = rows K+2,K+3
  // (row in lanes 0-15 / 16-31). h1b0..1 = K0..3, h1b2..3 = K4..7.
  float h1b0 = 0.f, h1b1 = 0.f, h1b2 = 0.f, h1b3 = 0.f;
  float h2b0 = 0.f, h2b1 = 0.f, h2b2 = 0.f, h2b3 = 0.f;
  float g0=0, g1=0, g2=0, g3=0, g4=0, g5=0, g6=0, g7=0;  // last layer-2 tanh

  // one float4 = two timesteps of this lane's sequence (16B aligned)
  const float4* xq = (const float4*)(x + (size_t)(b0 + m) * TT * II);
  float4 xv = xq[0];

  auto step = [&](float xk0, float xk1) {
    // X^T as B operand: row K=0 = x[:,0] (lanes 0-15), K=1 = x[:,1] (16-31)
    v2f xb; xb.x = hi ? xk1 : xk0; xb.y = 0.f;

    // ---- layer 0: acc = bias + W_ih0 @ X^T + W_hh0 @ H1^T ----
    v8f acc = wmma4(a_ih0, xb, cb0);
    v2f hA; hA.x = h1b0; hA.y = h1b1;
    v2f hB; hB.x = h1b2; hB.y = h1b3;
    acc = wmma4(a_hh0a, hA, acc);
    acc = wmma4(a_hh0b, hB, acc);

    float t0 = tanh_fast(acc[0]), t1 = tanh_fast(acc[1]);
    float t2 = tanh_fast(acc[2]), t3 = tanh_fast(acc[3]);
    float t4 = tanh_fast(acc[4]), t5 = tanh_fast(acc[5]);
    float t6 = tanh_fast(acc[6]), t7 = tanh_fast(acc[7]);
    // D -> B relayout: odd rows move to upper lane half (pure VALU)
    float s1 = xor16(t1), s3 = xor16(t3), s5 = xor16(t5), s7 = xor16(t7);
    h1b0 = hi ? s1 : t0;
    h1b1 = hi ? s3 : t2;
    h1b2 = hi ? s5 : t4;
    h1b3 = hi ? s7 : t6;

    // ---- layer 1: acc2 = bias + W_ih1 @ H1^T + W_hh1 @ H2^T ----
    v2f h1A; h1A.x = h1b0; h1A.y = h1b1;
    v2f h1B; h1B.x = h1b2; h1B.y = h1b3;
    v8f acc2 = wmma4(a_ih1a, h1A, cb1);
    acc2 = wmma4(a_ih1b, h1B, acc2);
    v2f h2A; h2A.x = h2b0; h2A.y = h2b1;
    v2f h2B; h2B.x = h2b2; h2B.y = h2b3;
    acc2 = wmma4(a_hh1a, h2A, acc2);
    acc2 = wmma4(a_hh1b, h2B, acc2);

    g0 = tanh_fast(acc2[0]); g1 = tanh_fast(acc2[1]);
    g2 = tanh_fast(acc2[2]); g3 = tanh_fast(acc2[3]);
    g4 = tanh_fast(acc2[4]); g5 = tanh_fast(acc2[5]);
    g6 = tanh_fast(acc2[6]); g7 = tanh_fast(acc2[7]);
    float u1 = xor16(g1), u3 = xor16(g3), u5 = xor16(g5), u7 = xor16(g7);
    h2b0 = hi ? u1 : g0;
    h2b1 = hi ? u3 : g2;
    h2b2 = hi ? u5 : g4;
    h2b3 = hi ? u7 : g6;
  };

  constexpr int TP = TT / 2;  // two timesteps per float4
  for (int tp = 0; tp < TP; ++tp) {
    const int tn = (tp + 1 < TP) ? (tp + 1) : tp;  // uniform clamp
    float4 xn = xq[tn];
    step(xv.x, xv.y);
    step(xv.z, xv.w);
    xv = xn;
  }

  // ---- head: sigmoid(h2_last . w_out + b_out), lanes 0-15 = batch ----
  if (hi == 0) {
    float z = b_out[0];
    z += g0 * w_out[0] + g1 * w_out[1] + g2 * w_out[2] + g3 * w_out[3];
    z += g4 * w_out[4] + g5 * w_out[5] + g6 * w_out[6] + g7 * w_out[7];
    out[b0 + m] = sigmoid_fast(z);
  }
}

extern "C" void kernel_launch(void* const* d_in, const int* in_sizes, int n_in,
                              void* d_out, int out_size, void* d_ws, size_t ws_size,
                              hipStream_t stream) {
  (void)in_sizes; (void)n_in; (void)out_size; (void)d_ws; (void)ws_size;
  const float* x     = (const float*)d_in[0];
  const float* w_ih0 = (const float*)d_in[1];
  const float* w_hh0 = (const float*)d_in[2];
  const float* b_ih0 = (const float*)d_in[3];
  const float* b_hh0 = (const float*)d_in[4];
  const float* w_ih1 = (const float*)d_in[5];
  const float* w_hh1 = (const float*)d_in[6];
  const float* b_ih1 = (const float*)d_in[7];
  const float* b_hh1 = (const float*)d_in[8];
  const float* w_out = (const float*)d_in[9];
  const float* b_out = (const float*)d_in[10];
  float* out = (float*)d_out;

  // 2048 batches / 16 per wave = 128 waves; 4 waves per block -> 32 blocks.
  dim3 grid(BB / (16 * 4));
  dim3 block(128);
  hipLaunchKernelGGL(rnn2_fused, grid, block, 0, stream,
                     x, w_ih0, w_hh0, b_ih0, b_hh0,
                     w_ih1, w_hh1, b_ih1, b_hh1, w_out, b_out, out);
}